// PatchNetVLAD_4312147165385
// MI455X (gfx1250) — compile-verified
//
#include <hip/hip_runtime.h>
#include <math.h>

typedef __attribute__((ext_vector_type(2))) float v2f;
typedef __attribute__((ext_vector_type(8))) float v8f;

#define N_   4
#define C_   128
#define K_   64
#define H_   30
#define W_   40
#define HW_  1200
#define HP_  27
#define WP_  37
#define NP_  999      /* 27*37 windows */
#define EPS_ 1e-12f

/* workspace layout (floats) */
#define XN_OFF 0
#define SA_OFF (N_*C_*HW_)            /* 614400 */
#define S1_OFF (SA_OFF + N_*K_*HW_)   /* 921600 */
#define S0_OFF (S1_OFF + N_*K_*C_)    /* 954368 */

/* ---- Stage 1: per-pixel L2 norm over channels ---- */
__global__ void k_l2norm_x(const float* __restrict__ x, float* __restrict__ xn) {
  int n = blockIdx.x / HW_;
  int p = blockIdx.x % HW_;
  int c = threadIdx.x;
  float v = x[(n * C_ + c) * HW_ + p];
  __shared__ float red[C_];
  red[c] = v * v;
  __syncthreads();
  for (int s = C_ / 2; s > 0; s >>= 1) {
    if (c < s) red[c] += red[c + s];
    __syncthreads();
  }
  float inv = 1.0f / fmaxf(sqrtf(red[0]), EPS_);
  xn[(n * C_ + c) * HW_ + p] = v * inv;
}

/* ---- Stage 2: logits + softmax over K per pixel ---- */
__global__ void k_softassign(const float* __restrict__ xn,
                             const float* __restrict__ conv_w,
                             float* __restrict__ sa) {
  int n = blockIdx.x / HW_;
  int p = blockIdx.x % HW_;
  int k = threadIdx.x;
  const float* wrow = conv_w + k * C_;
  const float* xcol = xn + (size_t)n * C_ * HW_ + p;
  float acc = 0.f;
#pragma unroll 8
  for (int c = 0; c < C_; ++c) acc += wrow[c] * xcol[c * HW_];
  __shared__ float red[K_];
  red[k] = acc;
  __syncthreads();
  for (int s = K_ / 2; s > 0; s >>= 1) {
    if (k < s) red[k] = fmaxf(red[k], red[k + s]);
    __syncthreads();
  }
  float m = red[0];
  __syncthreads();
  float e = expf(acc - m);
  red[k] = e;
  __syncthreads();
  for (int s = K_ / 2; s > 0; s >>= 1) {
    if (k < s) red[k] += red[k + s];
    __syncthreads();
  }
  sa[(n * K_ + k) * HW_ + p] = e / red[0];
}

/* ---- Stage 3: S1[n,k,c] = sum_p sa*xn ; S0[n,k] = sum_p sa ---- */
__global__ void k_s1(const float* __restrict__ xn, const float* __restrict__ sa,
                     float* __restrict__ S1, float* __restrict__ S0) {
  int n = blockIdx.x / K_;
  int k = blockIdx.x % K_;
  int c = threadIdx.x;
  const float* sar = sa + (size_t)(n * K_ + k) * HW_;
  const float* xr  = xn + (size_t)(n * C_ + c) * HW_;
  float acc = 0.f, asum = 0.f;
  for (int p = 0; p < HW_; ++p) {
    float s = sar[p];
    acc += s * xr[p];
    asum += s;
  }
  S1[(n * K_ + k) * C_ + c] = acc;
  if (c == 0) S0[n * K_ + k] = asum;
}

/* ---- Stage 4: vlad_global with two-level L2 norm ---- */
__global__ void k_vlad_global(const float* __restrict__ S1, const float* __restrict__ S0,
                              const float* __restrict__ cent, float* __restrict__ out) {
  int n = blockIdx.x;
  int tid = threadIdx.x;
  __shared__ float vals[K_ * 129];
  __shared__ float part[256];
  __shared__ float contrib[K_];
  __shared__ float invk[K_];
  __shared__ float s_inv_tot;
  for (int j = 0; j < 32; ++j) {
    int idx = tid + 256 * j;
    int k = idx >> 7, c = idx & 127;
    float r = S1[(n * K_ + k) * C_ + c] - cent[k * C_ + c] * S0[n * K_ + k];
    vals[k * 129 + c] = r;
  }
  __syncthreads();
  {
    int k = tid >> 2, r = tid & 3;
    float ss = 0.f;
    for (int c = r; c < C_; c += 4) { float v = vals[k * 129 + c]; ss += v * v; }
    part[tid] = ss;
  }
  __syncthreads();
  if (tid < K_) {
    float ss = part[4 * tid] + part[4 * tid + 1] + part[4 * tid + 2] + part[4 * tid + 3];
    float inv = 1.0f / fmaxf(sqrtf(ss), EPS_);
    invk[tid] = inv;
    contrib[tid] = ss * inv * inv;
  }
  __syncthreads();
  if (tid == 0) {
    float tot = 0.f;
    for (int k = 0; k < K_; ++k) tot += contrib[k];
    s_inv_tot = 1.0f / fmaxf(sqrtf(tot), EPS_);
  }
  __syncthreads();
  for (int j = 0; j < 32; ++j) {
    int idx = tid + 256 * j;
    int k = idx >> 7, c = idx & 127;
    out[n * (K_ * C_) + idx] = vals[k * 129 + c] * invk[k] * s_inv_tot;
  }
}

/* ---- Stage 5: per-window 64x128x16 GEMM via V_WMMA_F32_16X16X4_F32,
       then per-k and total L2 norms, write vlad_local ---- */
__global__ void __launch_bounds__(256)
k_vlad_local(const float* __restrict__ xn, const float* __restrict__ sa,
             const float* __restrict__ cent, float* __restrict__ out) {
  const int n  = blockIdx.x / NP_;
  const int pp = blockIdx.x % NP_;
  const int hp = pp / WP_, wp = pp % WP_;
  const int tid = threadIdx.x;
  const int lane = tid & 31, wave = tid >> 5;
  const int l16 = lane & 15, hi = lane >> 4;

  __shared__ float As[K_ * 17];    /* sa window, 64x16, padded   */
  __shared__ float Xs[C_ * 17];    /* xn window, 128x16, padded  */
  __shared__ float Ms[K_ * 129];   /* 64x128 result tile, padded */
  __shared__ float Pk[K_];
  __shared__ float part[256];
  __shared__ float contrib[K_];
  __shared__ float invk[K_];
  __shared__ float s_inv_tot;

  /* load A (sa window): 1024 elems, 4 per thread */
  const float* sab = sa + (size_t)n * K_ * HW_;
#pragma unroll
  for (int j = 0; j < 4; ++j) {
    int idx = tid + 256 * j;
    int k = idx >> 4, q = idx & 15;
    int r = hp + (q >> 2), cc = wp + (q & 3);
    As[k * 17 + q] = sab[k * HW_ + r * W_ + cc];
  }
  /* load X (xn window): 2048 elems, 8 per thread */
  const float* xb = xn + (size_t)n * C_ * HW_;
#pragma unroll
  for (int j = 0; j < 8; ++j) {
    int idx = tid + 256 * j;
    int c = idx >> 4, q = idx & 15;
    int r = hp + (q >> 2), cc = wp + (q & 3);
    Xs[c * 17 + q] = xb[c * HW_ + r * W_ + cc];
  }
  __syncthreads();

  /* box-sum of sa per k (window sum) */
  if (tid < K_) {
    float s = 0.f;
#pragma unroll
    for (int q = 0; q < 16; ++q) s += As[tid * 17 + q];
    Pk[tid] = s;
  }

  /* WMMA: 32 output tiles (4 k-tiles x 8 c-tiles), 4 per wave,
     each tile is D(16x16) += A(16x4) * B(4x16) over 4 K-steps */
#pragma unroll
  for (int i = 0; i < 4; ++i) {
    int t = wave * 4 + i;
    int kb = (t >> 3) * 16, cb = (t & 7) * 16;
    v8f acc = {0.f, 0.f, 0.f, 0.f, 0.f, 0.f, 0.f, 0.f};
#pragma unroll
    for (int s = 0; s < 4; ++s) {
      v2f a, b;
      /* A 16x4 layout: VGPR v holds K = v + 2*hi (ISA 7.12.2) */
      a.x = As[(kb + l16) * 17 + s * 4 + 2 * hi];
      a.y = As[(kb + l16) * 17 + s * 4 + 2 * hi + 1];
      /* B 4x16: row K = v + 2*hi striped across lanes (col = l16); B[q][c] = Xs[c][q] */
      b.x = Xs[(cb + l16) * 17 + s * 4 + 2 * hi];
      b.y = Xs[(cb + l16) * 17 + s * 4 + 2 * hi + 1];
      acc = __builtin_amdgcn_wmma_f32_16x16x4_f32(false, a, false, b,
                                                  (short)0, acc, false, false);
    }
    /* D layout: VGPR v -> row kb + v + 8*hi, col cb + l16 */
#pragma unroll
    for (int v = 0; v < 8; ++v)
      Ms[(kb + v + 8 * hi) * 129 + cb + l16] = acc[v];
  }
  __syncthreads();

  /* vl = (M - cent*Pk) / (ps*ps), overwrite Ms */
  for (int j = 0; j < 32; ++j) {
    int idx = tid + 256 * j;
    int k = idx >> 7, c = idx & 127;
    Ms[k * 129 + c] = (Ms[k * 129 + c] - cent[k * C_ + c] * Pk[k]) * 0.0625f;
  }
  __syncthreads();

  /* per-k sumsq over c */
  {
    int k = tid >> 2, r = tid & 3;
    float ss = 0.f;
    for (int c = r; c < C_; c += 4) { float v = Ms[k * 129 + c]; ss += v * v; }
    part[tid] = ss;
  }
  __syncthreads();
  if (tid < K_) {
    float ss = part[4 * tid] + part[4 * tid + 1] + part[4 * tid + 2] + part[4 * tid + 3];
    float inv = 1.0f / fmaxf(sqrtf(ss), EPS_);
    invk[tid] = inv;
    contrib[tid] = ss * inv * inv;
  }
  __syncthreads();
  if (tid == 0) {
    float tot = 0.f;
    for (int k = 0; k < K_; ++k) tot += contrib[k];
    s_inv_tot = 1.0f / fmaxf(sqrtf(tot), EPS_);
  }
  __syncthreads();

  /* write vlad_local[n, kc, pp] */
  float* ob = out + (size_t)N_ * K_ * C_ + (size_t)n * (K_ * C_) * NP_ + pp;
  for (int j = 0; j < 32; ++j) {
    int idx = tid + 256 * j;
    int k = idx >> 7, c = idx & 127;
    ob[(size_t)idx * NP_] = Ms[k * 129 + c] * invk[k] * s_inv_tot;
  }
}

extern "C" void kernel_launch(void* const* d_in, const int* in_sizes, int n_in,
                              void* d_out, int out_size, void* d_ws, size_t ws_size,
                              hipStream_t stream) {
  (void)in_sizes; (void)n_in; (void)out_size; (void)ws_size;
  const float* x     = (const float*)d_in[0];
  const float* convw = (const float*)d_in[1];
  const float* cent  = (const float*)d_in[2];
  float* out = (float*)d_out;
  float* ws  = (float*)d_ws;
  float* xn = ws + XN_OFF;
  float* sa = ws + SA_OFF;
  float* S1 = ws + S1_OFF;
  float* S0 = ws + S0_OFF;

  k_l2norm_x  <<<N_ * HW_, C_, 0, stream>>>(x, xn);
  k_softassign<<<N_ * HW_, K_, 0, stream>>>(xn, convw, sa);
  k_s1        <<<N_ * K_,  C_, 0, stream>>>(xn, sa, S1, S0);
  k_vlad_global<<<N_, 256, 0, stream>>>(S1, S0, cent, out);
  k_vlad_local <<<N_ * NP_, 256, 0, stream>>>(xn, sa, cent, out);
}